// HypergraphLayer_62732292325944
// MI455X (gfx1250) — compile-verified
//
#include <hip/hip_runtime.h>
#include <hip/hip_bf16.h>

typedef __attribute__((ext_vector_type(2))) float v2f;
typedef __attribute__((ext_vector_type(8))) float v8f;

#define N_NODES   50000
#define N_EDGES   800000
#define DIM       64
#define N_REL     5
#define NEG_SLOPE 0.01f

// ---------- order-preserving float <-> uint for atomicMax ----------
__device__ __forceinline__ unsigned f2ord(float f) {
    unsigned u = __float_as_uint(f);
    return (u & 0x80000000u) ? ~u : (u | 0x80000000u);
}
__device__ __forceinline__ float ord2f(unsigned k) {
    unsigned u = (k & 0x80000000u) ? (k & 0x7FFFFFFFu) : ~k;
    return __uint_as_float(u);
}

// ---------- tiny init ----------
__global__ void k_init(int* cnt, unsigned* keys) {
    int t = threadIdx.x;
    if (t < N_REL) cnt[t] = 0;
    if (t < 3 * DIM) keys[t] = 0x007FFFFFu;   // encode(-inf)
}

// ---------- edge norm + relation histogram ----------
__global__ void k_enorm_hist(const float* __restrict__ nn, const float* __restrict__ en,
                             const int* __restrict__ src, const int* __restrict__ dst,
                             const int* __restrict__ et, float* __restrict__ enorm,
                             int* __restrict__ cnt) {
    int i = blockIdx.x * blockDim.x + threadIdx.x;
    if (i >= N_EDGES) return;
    enorm[i] = nn[src[i]] * nn[dst[i]] * en[i];
    atomicAdd(&cnt[et[i]], 1);
}

// ---------- exclusive scan over 5 relations (trivial) ----------
__global__ void k_scan(const int* __restrict__ cnt, int* __restrict__ off,
                       int* __restrict__ gbase, int* __restrict__ fill) {
    off[0] = 0; gbase[0] = 0;
    for (int r = 0; r < N_REL; ++r) {
        off[r + 1]   = off[r] + cnt[r];
        gbase[r + 1] = gbase[r] + (cnt[r] + 63) / 64;   // 64-edge groups
        fill[r]      = off[r];
    }
}

// ---------- bucket edges by relation ----------
__global__ void k_scatter(const int* __restrict__ et, int* __restrict__ fill,
                          int* __restrict__ bucket) {
    int i = blockIdx.x * blockDim.x + threadIdx.x;
    if (i >= N_EDGES) return;
    int p = atomicAdd(&fill[et[i]], 1);
    bucket[p] = i;
}

// ---------- zero a float buffer ----------
__global__ void k_zero(float* __restrict__ p, int n) {
    for (int i = blockIdx.x * blockDim.x + threadIdx.x; i < n; i += gridDim.x * blockDim.x)
        p[i] = 0.f;
}

// ---------- column max (64 cols) ----------
__global__ void k_colmax(const float* __restrict__ h, unsigned* __restrict__ keys, int nrows) {
    int col = threadIdx.x & 63;
    int rpb = blockDim.x >> 6;
    float m = -__builtin_inff();
    for (int row = blockIdx.x * rpb + (threadIdx.x >> 6); row < nrows; row += gridDim.x * rpb)
        m = fmaxf(m, h[row * DIM + col]);
    atomicMax(&keys[col], f2ord(m));
}

// ---------- main grouped-GEMM message kernel ----------
// 128 threads = 4 waves; wave w computes output columns [16w, 16w+16).
// Each block owns one 64-edge group of a single relation: B fragments
// (64x16 slices of W_src[r], W_aff[r]) loaded once, reused for 4 M=16 tiles.
__global__ __launch_bounds__(128) void k_rgemm(
    const float* __restrict__ h_cur,
    const float* __restrict__ Wsrc_l,   // [R][64][64] for this layer
    const float* __restrict__ Waff_l,
    const int* __restrict__ src, const int* __restrict__ dst,
    const float* __restrict__ enorm,
    const int* __restrict__ bucket,
    const int* __restrict__ off,        // R+1 edge offsets
    const int* __restrict__ gbase,      // R+1 group offsets
    float* __restrict__ h_next) {

    __shared__ float hs[16 * 68];       // gathered h[src] rows (pad 68: bank-conflict free)
    __shared__ float pr[16 * 68];       // h[src]*h[dst] rows
    __shared__ int   dsts[16];
    __shared__ float ens[16];

    const int b = blockIdx.x;
    if (b >= gbase[N_REL]) return;
    int r = 0;
    while (r < N_REL - 1 && b >= gbase[r + 1]) ++r;
    const int g    = b - gbase[r];
    const int rbeg = off[r];
    const int rcnt = off[r + 1] - rbeg;

    const int tid  = threadIdx.x;
    const int wv   = tid >> 5;
    const int lane = tid & 31;
    const int n    = (wv << 4) + (lane & 15);   // output column
    const int kh   = (lane >> 4) << 1;          // 0 (lanes 0-15) / 2 (lanes 16-31)
    const int mrow = lane & 15;                 // A-matrix row

    // B fragments: per ISA 16x4 layout, VGPR0 = K{0|2}, VGPR1 = K{1|3}
    const float* Ws = Wsrc_l + r * DIM * DIM;
    const float* Wa = Waff_l + r * DIM * DIM;
    v2f bs[16], ba[16];
#pragma unroll
    for (int kc = 0; kc < 16; ++kc) {
        int k0 = 4 * kc + kh;
        bs[kc].x = Ws[k0 * DIM + n];  bs[kc].y = Ws[(k0 + 1) * DIM + n];
        ba[kc].x = Wa[k0 * DIM + n];  ba[kc].y = Wa[(k0 + 1) * DIM + n];
    }

    for (int t = 0; t < 4; ++t) {
        const int ebase = g * 64 + t * 16;      // relative to rbeg
        const int rem   = rcnt - ebase;
        if (rem <= 0) break;                    // uniform across block
        const int cnt = rem < 16 ? rem : 16;

        __syncthreads();                        // protect LDS reuse
        for (int idx = tid; idx < 16 * DIM; idx += 128) {
            int m = idx >> 6, k = idx & 63;
            float a = 0.f, p = 0.f;
            if (m < cnt) {
                int e  = bucket[rbeg + ebase + m];
                int sN = src[e], dN = dst[e];
                a = h_cur[sN * DIM + k];
                p = a * h_cur[dN * DIM + k];
                if (k == 0) { dsts[m] = dN; ens[m] = enorm[e]; }
            } else if (k == 0) { dsts[m] = 0; ens[m] = 0.f; }
            hs[m * 68 + k] = a;
            pr[m * 68 + k] = p;
        }
        __syncthreads();

        v8f c = {0.f, 0.f, 0.f, 0.f, 0.f, 0.f, 0.f, 0.f};
#pragma unroll
        for (int kc = 0; kc < 16; ++kc) {
            int k0 = 4 * kc + kh;
            v2f a, p;
            a.x = hs[mrow * 68 + k0]; a.y = hs[mrow * 68 + k0 + 1];
            p.x = pr[mrow * 68 + k0]; p.y = pr[mrow * 68 + k0 + 1];
            c = __builtin_amdgcn_wmma_f32_16x16x4_f32(false, a, false, bs[kc],
                                                      (short)0, c, false, false);
            c = __builtin_amdgcn_wmma_f32_16x16x4_f32(false, p, false, ba[kc],
                                                      (short)0, c, false, false);
        }

        // C layout: VGPR i -> row i + 8*(lane>>4), col = lane&15
#pragma unroll
        for (int i = 0; i < 8; ++i) {
            int mr = i + ((lane >> 4) << 3);
            if (mr < cnt) {
                float v = c[i] * ens[mr];
                v = (v > 0.f) ? v : v * NEG_SLOPE;
                atomicAdd(&h_next[dsts[mr] * DIM + n], v);
            }
        }
    }
}

// ---------- finalize: mean of 3 column maxes ----------
__global__ void k_final(const unsigned* __restrict__ keys, float* __restrict__ out) {
    int d = threadIdx.x;
    if (d >= DIM) return;
    out[d] = (ord2f(keys[d]) + ord2f(keys[DIM + d]) + ord2f(keys[2 * DIM + d])) * (1.0f / 3.0f);
}

extern "C" void kernel_launch(void* const* d_in, const int* in_sizes, int n_in,
                              void* d_out, int out_size, void* d_ws, size_t ws_size,
                              hipStream_t stream) {
    const float* x   = (const float*)d_in[0];
    const float* nn  = (const float*)d_in[1];
    const float* en  = (const float*)d_in[2];
    const float* Wsr = (const float*)d_in[3];   // [L][R][64][64]
    const float* Waf = (const float*)d_in[4];
    const int*   src = (const int*)d_in[5];
    const int*   dst = (const int*)d_in[6];
    const int*   et  = (const int*)d_in[7];
    float* out = (float*)d_out;

    // workspace layout
    char* ws = (char*)d_ws;
    float* h1    = (float*)ws;  ws += sizeof(float) * N_NODES * DIM;   // 12.8 MB
    float* h2    = (float*)ws;  ws += sizeof(float) * N_NODES * DIM;   // 12.8 MB
    float* enorm = (float*)ws;  ws += sizeof(float) * N_EDGES;         // 3.2 MB
    int* bucket  = (int*)ws;    ws += sizeof(int) * N_EDGES;           // 3.2 MB
    int* cnt     = (int*)ws;    ws += sizeof(int) * N_REL;
    int* off     = (int*)ws;    ws += sizeof(int) * (N_REL + 1);
    int* gbase   = (int*)ws;    ws += sizeof(int) * (N_REL + 1);
    int* fill    = (int*)ws;    ws += sizeof(int) * N_REL;
    unsigned* keys = (unsigned*)ws;                                    // 3*64 uints

    const int eblocks = (N_EDGES + 255) / 256;
    const int ngroups = (N_EDGES + 63) / 64 + N_REL;   // upper bound on 64-edge groups
    const int WperL   = N_REL * DIM * DIM;

    k_init<<<1, 256, 0, stream>>>(cnt, keys);
    k_enorm_hist<<<eblocks, 256, 0, stream>>>(nn, en, src, dst, et, enorm, cnt);
    k_scan<<<1, 1, 0, stream>>>(cnt, off, gbase, fill);
    k_scatter<<<eblocks, 256, 0, stream>>>(et, fill, bucket);
    k_zero<<<2048, 256, 0, stream>>>(h1, N_NODES * DIM);
    k_zero<<<2048, 256, 0, stream>>>(h2, N_NODES * DIM);

    k_colmax<<<512, 256, 0, stream>>>(x, keys, N_NODES);               // outs[0]

    k_rgemm<<<ngroups, 128, 0, stream>>>(x, Wsr, Waf, src, dst, enorm,
                                         bucket, off, gbase, h1);      // layer 0
    k_colmax<<<512, 256, 0, stream>>>(h1, keys + DIM, N_NODES);        // outs[1]

    k_rgemm<<<ngroups, 128, 0, stream>>>(h1, Wsr + WperL, Waf + WperL, src, dst, enorm,
                                         bucket, off, gbase, h2);      // layer 1
    k_colmax<<<512, 256, 0, stream>>>(h2, keys + 2 * DIM, N_NODES);    // outs[2]

    k_final<<<1, 64, 0, stream>>>(keys, out);
}